// Model_72748156060318
// MI455X (gfx1250) — compile-verified
//
#include <hip/hip_runtime.h>
#include <math.h>

// ---------------- problem constants ----------------
constexpr int B    = 128;
constexpr int E    = 14541;
constexpr int R    = 237;
constexpr int NREL = 2 * R;       // 474
constexpr int NT   = 272115;
constexpr int T    = 3;
constexpr int D    = 128;
constexpr int G4   = 4 * D;       // 512
constexpr float TAU1 = 10.0f;
constexpr float EPS  = 1e-9f;

typedef __attribute__((ext_vector_type(2))) float v2f;
typedef __attribute__((ext_vector_type(8))) float v8f;

__device__ __forceinline__ float sigmf(float x) { return 1.0f / (1.0f + __expf(-x)); }

// ---------------- utility ----------------
__global__ void zero_f32(float* __restrict__ p, int n) {
    int i = blockIdx.x * blockDim.x + threadIdx.x;
    if (i < n) p[i] = 0.0f;
}

// xq[b][d] = emb[input_r[b]][d];  xend[b][d] = emb[NREL][d]
__global__ void embed_q(const float* __restrict__ emb, const int* __restrict__ input_r,
                        float* __restrict__ xq, float* __restrict__ xend) {
    int idx = blockIdx.x * blockDim.x + threadIdx.x;   // B*D
    int b = idx >> 7, d = idx & 127;
    xq[idx]   = emb[input_r[b] * D + d];
    xend[idx] = emb[NREL * D + d];
}

// ---------------- LSTM gates: g = x@Wih^T + h@Whh^T + bih + bhh --------------
// 256 waves, one 16x16 tile each (8 row-tiles x 32 col-tiles), K stepped by 4
// via V_WMMA_F32_16X16X4_F32 (FP32 in/out, full precision for the recurrence).
__global__ void gates_wmma(const float* __restrict__ x, const float* __restrict__ h,
                           const float* __restrict__ Wih, const float* __restrict__ Whh,
                           const float* __restrict__ bih, const float* __restrict__ bhh,
                           float* __restrict__ g) {
    int wave = (blockIdx.x * blockDim.x + threadIdx.x) >> 5;   // 0..255
    int lane = threadIdx.x & 31;
    int tm = wave & 7, tn = wave >> 3;
    int row0 = tm * 16, col0 = tn * 16;
    int half = lane >> 4, mr = lane & 15;
    int col = col0 + mr;
    v8f acc = {};
    // A: 16x4 fragment, lane half selects K pair (ISA 7.12.2); B mirrors it.
    const float* arow = x + (row0 + mr) * D;
    const float* brow = Wih + col * D;           // B[k][j] = Wih[j][k]
    for (int k0 = 0; k0 < D; k0 += 4) {
        v2f a, bf;
        a.x  = arow[k0 + 2 * half];     a.y  = arow[k0 + 2 * half + 1];
        bf.x = brow[k0 + 2 * half];     bf.y = brow[k0 + 2 * half + 1];
        acc = __builtin_amdgcn_wmma_f32_16x16x4_f32(false, a, false, bf,
                                                    (short)0, acc, false, false);
    }
    arow = h + (row0 + mr) * D;
    brow = Whh + col * D;
    for (int k0 = 0; k0 < D; k0 += 4) {
        v2f a, bf;
        a.x  = arow[k0 + 2 * half];     a.y  = arow[k0 + 2 * half + 1];
        bf.x = brow[k0 + 2 * half];     bf.y = brow[k0 + 2 * half + 1];
        acc = __builtin_amdgcn_wmma_f32_16x16x4_f32(false, a, false, bf,
                                                    (short)0, acc, false, false);
    }
    float bias = bih[col] + bhh[col];
    for (int v = 0; v < 8; ++v)                         // C/D: M = v + 8*half, N = mr
        g[(row0 + v + 8 * half) * G4 + col] = acc[v] + bias;
}

// c = sig(f)*c + sig(i)*tanh(g); h = sig(o)*tanh(c)
__global__ void lstm_elem(const float* __restrict__ g, float* __restrict__ c,
                          float* __restrict__ hout) {
    int idx = blockIdx.x * blockDim.x + threadIdx.x;   // B*D
    int b = idx >> 7, d = idx & 127;
    const float* gr = g + b * G4;
    float gi = gr[d], gf = gr[D + d], gg = gr[2 * D + d], go = gr[3 * D + d];
    float cc = sigmf(gf) * c[idx] + sigmf(gi) * tanhf(gg);
    c[idx] = cc;
    hout[idx] = sigmf(go) * tanhf(cc);
}

// ---------------- w_all = rnn[:3B] @ Wlin^T + blin (384 x 474) ---------------
__global__ void wlin_wmma(const float* __restrict__ A, const float* __restrict__ Wlin,
                          const float* __restrict__ blin, float* __restrict__ wall) {
    int wave = (blockIdx.x * blockDim.x + threadIdx.x) >> 5;   // 0..719
    int lane = threadIdx.x & 31;
    int tm = wave % 24, tn = wave / 24;                // 24 row tiles, 30 col tiles
    int row0 = tm * 16, col0 = tn * 16;
    int half = lane >> 4, mr = lane & 15;
    int col = col0 + mr;
    bool cok = col < NREL;
    v8f acc = {};
    const float* arow = A + (row0 + mr) * D;
    const float* brow = Wlin + (cok ? col : 0) * D;
    for (int k0 = 0; k0 < D; k0 += 4) {
        v2f a, bf;
        a.x = arow[k0 + 2 * half];  a.y = arow[k0 + 2 * half + 1];
        float b0 = brow[k0 + 2 * half], b1 = brow[k0 + 2 * half + 1];
        bf.x = cok ? b0 : 0.0f;     bf.y = cok ? b1 : 0.0f;
        acc = __builtin_amdgcn_wmma_f32_16x16x4_f32(false, a, false, bf,
                                                    (short)0, acc, false, false);
    }
    if (cok) {
        float bias = blin[col];
        for (int v = 0; v < 8; ++v)
            wall[(row0 + v + 8 * half) * NREL + col] = acc[v] + bias;
    }
}

// softmax over NREL per (t,b) row; write transposed: wsoft[t][r*B + b]
__global__ void softmax_w(const float* __restrict__ wall, float* __restrict__ wsoft) {
    int row = blockIdx.x * blockDim.x + threadIdx.x;   // 0..383 (= t*B + b)
    if (row >= T * B) return;
    int t = row / B, b = row % B;
    const float* src = wall + (size_t)row * NREL;
    float mx = -1e30f;
    for (int r = 0; r < NREL; ++r) mx = fmaxf(mx, src[r]);
    float s = 0.0f;
    for (int r = 0; r < NREL; ++r) s += __expf(src[r] - mx);
    float inv = 1.0f / s;
    float* dst = wsoft + (size_t)t * NREL * B;
    for (int r = 0; r < NREL; ++r) dst[r * B + b] = __expf(src[r] - mx) * inv;
}

__global__ void onehot_k(const int* __restrict__ input_x, float* __restrict__ s0) {
    int b = threadIdx.x;                               // 128
    s0[(size_t)input_x[b] * B + b] = 1.0f;
}

// att[b][tau] = softmax_tau( rnn[t][b] . rnn[tau][b] )
__global__ void att_kernel(const float* __restrict__ rnn, float* __restrict__ att, int t) {
    int b = threadIdx.x;                               // 128
    float sc[4]; float mx = -1e30f;
    const float* at = rnn + ((size_t)t * B + b) * D;
    for (int tau = 0; tau <= t; ++tau) {
        const float* rt = rnn + ((size_t)tau * B + b) * D;
        float d = 0.0f;
        for (int k = 0; k < D; ++k) d += at[k] * rt[k];
        sc[tau] = d; mx = fmaxf(mx, d);
    }
    float s = 0.0f;
    for (int tau = 0; tau <= t; ++tau) { sc[tau] = __expf(sc[tau] - mx); s += sc[tau]; }
    for (int tau = 0; tau <= t; ++tau) att[b * 4 + tau] = sc[tau] / s;
}

// inp[e*B+b] = sum_tau att[b][tau] * states[tau][e*B+b]   (layout (E,B))
__global__ void inp_kernel(const float* __restrict__ att,
                           const float* __restrict__ s0, const float* __restrict__ s1,
                           const float* __restrict__ s2, const float* __restrict__ s3,
                           float* __restrict__ inp, int nt) {
    int idx = blockIdx.x * blockDim.x + threadIdx.x;
    if (idx >= E * B) return;
    int b = idx & 127;
    float v = att[b * 4 + 0] * s0[idx];
    if (nt > 1) v += att[b * 4 + 1] * s1[idx];
    if (nt > 2) v += att[b * 4 + 2] * s2[idx];
    if (nt > 3) v += att[b * 4 + 3] * s3[idx];
    inp[idx] = v;
}

// scatter messages: fully coalesced (E,B)/(NREL,B) layout; native f32 atomics
__global__ void edge_kernel(const int* __restrict__ th, const int* __restrict__ tt,
                            const int* __restrict__ tr, const float* __restrict__ inp,
                            const float* __restrict__ w, float* __restrict__ sout) {
    int idx = blockIdx.x * blockDim.x + threadIdx.x;   // NT*B = 34.8M < 2^31
    if (idx >= NT * B) return;
    int e = idx >> 7, b = idx & 127;
    int hh = th[e], tl = tt[e], r = tr[e];
    float mf = inp[(size_t)hh * B + b] * w[(size_t)r * B + b];
    float mb = inp[(size_t)tl * B + b] * w[(size_t)(r + R) * B + b];
    unsafeAtomicAdd(&sout[(size_t)tl * B + b], mf);
    unsafeAtomicAdd(&sout[(size_t)hh * B + b], mb);
}

// clamp to TAU1 in place + accumulate row (per-b) sums
__global__ void clamp_sum(float* __restrict__ s, float* __restrict__ sums) {
    int b = threadIdx.x;                               // 128
    int e0 = blockIdx.x * 128;
    float loc = 0.0f;
    for (int i = 0; i < 128; ++i) {
        int e = e0 + i;
        if (e >= E) break;
        float v = fminf(s[(size_t)e * B + b], TAU1);
        s[(size_t)e * B + b] = v;
        loc += v;
    }
    unsafeAtomicAdd(&sums[b], loc);
}

__global__ void normalize_k(float* __restrict__ s, const float* __restrict__ sums) {
    int idx = blockIdx.x * blockDim.x + threadIdx.x;
    if (idx >= E * B) return;
    int b = idx & 127;
    s[idx] = s[idx] / fmaxf(sums[b], EPS);
}

// (E,B) -> (B,E) tiled transpose through LDS
__global__ void transpose_out(const float* __restrict__ inp, float* __restrict__ out) {
    __shared__ float tile[32][33];
    int e0 = blockIdx.x * 32;
    int b0 = blockIdx.y * 32;
    int tx = threadIdx.x & 31;
    int ty4 = threadIdx.x >> 5;                        // 0..7
    for (int i = 0; i < 4; ++i) {
        int ty = ty4 + i * 8;
        int e = e0 + ty;
        if (e < E) tile[ty][tx] = inp[(size_t)e * B + (b0 + tx)];
    }
    __syncthreads();
    for (int i = 0; i < 4; ++i) {
        int ty = ty4 + i * 8;
        int e = e0 + tx;
        if (e < E) out[(size_t)(b0 + ty) * E + e] = tile[tx][ty];
    }
}

// ---------------- host driver ----------------
extern "C" void kernel_launch(void* const* d_in, const int* in_sizes, int n_in,
                              void* d_out, int out_size, void* d_ws, size_t ws_size,
                              hipStream_t stream) {
    (void)in_sizes; (void)n_in; (void)out_size; (void)ws_size;
    const int*   input_x  = (const int*)d_in[0];
    const int*   input_r  = (const int*)d_in[1];
    const int*   triple_h = (const int*)d_in[2];
    const int*   triple_t = (const int*)d_in[3];
    const int*   triple_r = (const int*)d_in[4];
    const float* emb      = (const float*)d_in[5];
    const float* Wih      = (const float*)d_in[6];
    const float* Whh      = (const float*)d_in[7];
    const float* bih      = (const float*)d_in[8];
    const float* bhh      = (const float*)d_in[9];
    const float* Wlin     = (const float*)d_in[10];
    const float* blin     = (const float*)d_in[11];
    float* out = (float*)d_out;

    float* w = (float*)d_ws;
    size_t off = 0;
    auto alloc = [&](size_t n) { float* p = w + off; off += n; return p; };
    float* rnn   = alloc((size_t)4 * B * D);        // h per step (t,B,D)
    float* xq    = alloc((size_t)B * D);
    float* xend  = alloc((size_t)B * D);
    float* cbuf  = alloc((size_t)B * D);
    float* hzero = alloc((size_t)B * D);            // contiguous with cbuf
    float* gbuf  = alloc((size_t)B * G4);
    float* wall  = alloc((size_t)T * B * NREL);
    float* wsoft = alloc((size_t)T * NREL * B);
    float* attb  = alloc((size_t)B * 4);
    float* sums  = alloc((size_t)B);
    float* st[4];
    for (int i = 0; i < 4; ++i) st[i] = alloc((size_t)E * B);
    float* inpb  = alloc((size_t)E * B);

    const int EBB = (E * B + 255) / 256;

    // h0 = c0 = 0 (cbuf and hzero contiguous)
    zero_f32<<<(2 * B * D + 255) / 256, 256, 0, stream>>>(cbuf, 2 * B * D);
    embed_q<<<(B * D + 255) / 256, 256, 0, stream>>>(emb, input_r, xq, xend);

    // 4-step LSTM (sequence = [q,q,q,end])
    for (int t = 0; t < 4; ++t) {
        const float* xt = (t < T) ? xq : xend;
        const float* ht = (t == 0) ? hzero : rnn + (size_t)(t - 1) * B * D;
        gates_wmma<<<64, 128, 0, stream>>>(xt, ht, Wih, Whh, bih, bhh, gbuf);
        lstm_elem<<<(B * D + 255) / 256, 256, 0, stream>>>(gbuf, cbuf,
                                                           rnn + (size_t)t * B * D);
    }

    // relation logits + softmax (written transposed per timestep)
    wlin_wmma<<<180, 128, 0, stream>>>(rnn, Wlin, blin, wall);
    softmax_w<<<3, 128, 0, stream>>>(wall, wsoft);

    // states[0] = one_hot(input_x)  in (E,B) layout
    zero_f32<<<EBB, 256, 0, stream>>>(st[0], E * B);
    onehot_k<<<1, 128, 0, stream>>>(input_x, st[0]);

    for (int t = 0; t < 4; ++t) {
        att_kernel<<<1, 128, 0, stream>>>(rnn, attb, t);
        inp_kernel<<<EBB, 256, 0, stream>>>(attb, st[0], st[1], st[2], st[3],
                                            inpb, t + 1);
        if (t < T) {
            zero_f32<<<EBB, 256, 0, stream>>>(st[t + 1], E * B);
            zero_f32<<<1, 128, 0, stream>>>(sums, B);
            edge_kernel<<<(NT * B + 255) / 256, 256, 0, stream>>>(
                triple_h, triple_t, triple_r, inpb,
                wsoft + (size_t)t * NREL * B, st[t + 1]);
            clamp_sum<<<(E + 127) / 128, 128, 0, stream>>>(st[t + 1], sums);
            normalize_k<<<EBB, 256, 0, stream>>>(st[t + 1], sums);
        } else {
            transpose_out<<<dim3((E + 31) / 32, B / 32), 256, 0, stream>>>(inpb, out);
        }
    }
}